// HashEmbedder_11106785427532
// MI455X (gfx1250) — compile-verified
//
#include <hip/hip_runtime.h>
#include <hip/hip_fp16.h>

#ifndef __has_builtin
#define __has_builtin(x) 0
#endif

constexpr int      kLevels      = 16;
constexpr unsigned kLog2T       = 19;
constexpr unsigned kT           = 1u << kLog2T;   // hash table entries per level
constexpr unsigned kMask        = kT - 1u;
constexpr unsigned kP1          = 2654435761u;    // instant-NGP primes (prime0 == 1)
constexpr unsigned kP2          = 805459861u;
constexpr int      kThreads     = 256;
constexpr int      kPtsPerBlock = 128;
constexpr int      kIters       = kPtsPerBlock * kLevels / kThreads; // 8

__global__ __launch_bounds__(kThreads)
void hashgrid_fwd(const float* __restrict__ points,        // [B,3] f32
                  const __half2* __restrict__ tables,      // [16 * 2^19] half2 (uniform base)
                  unsigned long long* __restrict__ out,    // [B,16] levelwise float2 packed as u64
                  int B)
{
  __shared__ float s_pts[kPtsPerBlock * 3];

  const int t      = (int)threadIdx.x;
  const int level  = t & (kLevels - 1);
  const int slot   = t >> 4;                      // 0..15: point slot within iteration
  const long long basePt = (long long)blockIdx.x * kPtsPerBlock;

  // ---- Stage this block's 128 points (384 dwords) into LDS via CDNA5 async loads ----
  {
    const int nDw = kPtsPerBlock * 3; // 384
    const long long lim = (long long)B * 3;
    #pragma unroll
    for (int d = t; d < nDw; d += kThreads) {
      const long long gDw = basePt * 3 + d;
      if (gDw < lim) {
#if __has_builtin(__builtin_amdgcn_global_load_async_to_lds_b32)
        auto g = (__attribute__((address_space(1))) int*)(points + gDw);
        auto l = (__attribute__((address_space(3))) int*)(&s_pts[d]);
        __builtin_amdgcn_global_load_async_to_lds_b32(g, l, /*offset=*/0, /*cpol=*/0);
#else
        s_pts[d] = points[gDw];
#endif
      }
    }
  }
#if __has_builtin(__builtin_amdgcn_s_wait_asynccnt)
  __builtin_amdgcn_s_wait_asynccnt(0);
#else
  asm volatile("s_wait_asynccnt 0" ::: "memory");
#endif
  __syncthreads();

  // Per-level constants (amortized over kIters points):
  //   res = floor(16 * 2^(level/3));  lvlBase = level * 2^19, folded into a 32-bit index
  //   so every gather is saddr(tables) + 32-bit voffset — no per-lane 64-bit math.
  const unsigned lvlBase = (unsigned)level << kLog2T;
  const float    res     = floorf(16.0f * __builtin_exp2f((float)level * 0.33333334f));

  // Uniform store base; per-lane offsets stay 32-bit.
  unsigned long long* __restrict__ outBlk = out + basePt * kLevels;
  const long long rem = (long long)B - basePt;
  const int nPts = rem < (long long)kPtsPerBlock ? (int)(rem < 0 ? 0 : rem) : kPtsPerBlock;

  #pragma unroll
  for (int it = 0; it < kIters; ++it) {
    const int lp = slot + it * (kThreads / kLevels);
    if (lp >= nPts) break;                        // 32-bit compare, uniform per wave-half

    const float rx = s_pts[lp * 3 + 0] * res;
    const float ry = s_pts[lp * 3 + 1] * res;
    const float rz = s_pts[lp * 3 + 2] * res;

    // floor / fractional part (rx >= 0, exact): u32 trunc + v_fract
    const unsigned bx = (unsigned)rx, by = (unsigned)ry, bz = (unsigned)rz;
    const float wx = rx - floorf(rx);
    const float wy = ry - floorf(ry);
    const float wz = rz - floorf(rz);

    // Spatial hash: idx = (x*1 ^ y*P1 ^ z*P2) & mask, level folded in via v_and_or
    const unsigned hy0 = by * kP1;
    const unsigned hz0 = bz * kP2;
    const unsigned hxy[4] = { bx ^ hy0,        bx ^ (hy0 + kP1),
                              (bx + 1u) ^ hy0, (bx + 1u) ^ (hy0 + kP1) };
    const unsigned hz[2]  = { hz0, hz0 + kP2 };

    __half2 e[8];
    #pragma unroll
    for (int ij = 0; ij < 4; ++ij)
      #pragma unroll
      for (int k = 0; k < 2; ++k)
        e[ij * 2 + k] = tables[((hxy[ij] ^ hz[k]) & kMask) | lvlBase];

    // Trilinear blend; fmaf(wc, (float)h16, acc) -> v_fma_mix_f32
    const float vx[2] = { 1.0f - wx, wx };
    const float vy[2] = { 1.0f - wy, wy };
    const float vz[2] = { 1.0f - wz, wz };
    float a0 = 0.0f, a1 = 0.0f;
    #pragma unroll
    for (int i = 0; i < 2; ++i) {
      #pragma unroll
      for (int j = 0; j < 2; ++j) {
        const float vxy = vx[i] * vy[j];
        #pragma unroll
        for (int k = 0; k < 2; ++k) {
          const float wc = vxy * vz[k];
          const __half2 h = e[(i * 2 + j) * 2 + k];
          a0 = fmaf(wc, __low2float(h),  a0);
          a1 = fmaf(wc, __high2float(h), a1);
        }
      }
    }

    // Coalesced (lane-consecutive) nontemporal b64 store: keep L2 for the tables.
    union { float f[2]; unsigned long long u; } pk;
    pk.f[0] = a0; pk.f[1] = a1;
    const unsigned oIdx = (unsigned)lp * (unsigned)kLevels + (unsigned)level;
    __builtin_nontemporal_store(pk.u, outBlk + oIdx);
  }
}

extern "C" void kernel_launch(void* const* d_in, const int* in_sizes, int n_in,
                              void* d_out, int out_size, void* d_ws, size_t ws_size,
                              hipStream_t stream) {
  const float*   points = (const float*)d_in[0];     // [B,3] f32
  const __half2* tables = (const __half2*)d_in[1];   // [16, 2^19, 2] f16 viewed as half2
  (void)n_in; (void)out_size; (void)d_ws; (void)ws_size;

  const int B = in_sizes[0] / 3;
  const int grid = (B + kPtsPerBlock - 1) / kPtsPerBlock;
  hashgrid_fwd<<<grid, kThreads, 0, stream>>>(points, tables,
                                              (unsigned long long*)d_out, B);
}